// dissected_Conv2d_30176440222031
// MI455X (gfx1250) — compile-verified
//
#include <hip/hip_runtime.h>

typedef __attribute__((ext_vector_type(2))) float v2f;
typedef __attribute__((ext_vector_type(8))) float v8f;

#define CIN   64
#define COUT  128
#define Hdim  56
#define Wdim  56
#define HWsz  (Hdim * Wdim)     // 3136
#define Bn    2
#define Npix  (Bn * HWsz)       // 6272 = 196 tiles of 32
#define KTOT  (CIN * 9)         // 576
#define NTB   32                // pixels (N) per block
#define NPAIR (KTOT / 2)        // 288 K-pairs
#define TILES_PER_BATCH (HWsz / NTB)       // 98: batch boundary == tile boundary
#define PACK_ELEMS ((size_t)COUT * KTOT)   // 73728 floats = 288 KB

// Repack weights [Cout][Cin][3][3] -> [Cout][tap][Cin]: K ordered (kh,kw,cin)
// becomes fully contiguous, so each per-lane A operand is one aligned
// global_load_b64 and the whole K loop is a single flat stream.
__global__ __launch_bounds__(256) void pack_weights(
    const float* __restrict__ wgt, float* __restrict__ wp)
{
  const int i   = blockIdx.x * 256 + threadIdx.x;  // destination index
  const int m   = i / KTOT;
  const int r   = i - m * KTOT;
  const int tap = r >> 6;      // 0..8
  const int cin = r & 63;
  wp[i] = wgt[m * KTOT + cin * 9 + tap];
}

// One block = 8 waves; block owns a 32-pixel N tile and full Cout=128.
// Phase 1: stage 576x32 im2col tile in LDS as K-pairs [pair][n][2] (72 KB);
//          pad bounds computed once per tap, zeros resolved at fill time.
// Phase 2: flat K loop, 144 steps/wave; per step 1 global b64 (A, shared by
//          both N sub-tiles) + 2 LDS b64 (B) + 2 x V_WMMA_F32_16X16X4_F32.
template <bool PACKED>
__global__ __launch_bounds__(256) void conv3x3_wmma(
    const float* __restrict__ x,     // [B, Cin, H, W]
    const float* __restrict__ wA,    // packed [Cout][tap][Cin] or raw weights
    const float* __restrict__ bias,  // [Cout]
    float* __restrict__ out)         // [B, Cout, H, W]
{
  __shared__ __align__(16) float lds[NPAIR * NTB * 2];  // 72 KB

  const int tid   = threadIdx.x;
  const int lane  = tid & 31;
  const int wave  = tid >> 5;
  const int col   = lane & 15;
  const int khalf = lane >> 4;

  // Batch index is uniform per block (tile boundary aligns with batch).
  const int bb  = (blockIdx.x >= TILES_PER_BATCH) ? 1 : 0;
  const int n0  = blockIdx.x * NTB;
  const int hw0 = n0 - bb * HWsz;        // tile start within this batch plane

  // ---------------- Phase 1: im2col fill ------------------------------------
  // Thread owns pixel fn = tid&31 and cin lanes {wv, wv+8, ..., wv+56}.
  {
    const int fn = tid & 31;
    const int wv = tid >> 5;
    const int hw = hw0 + fn;
    const int h  = hw / Wdim;
    const int w  = hw - h * Wdim;
    const float* xb = x + (size_t)bb * (CIN * HWsz);

    #pragma unroll
    for (int tap = 0; tap < 9; ++tap) {
      const int  kh = tap / 3;           // constants after unroll
      const int  kw = tap - kh * 3;
      const int  y  = h + kh - 1;
      const int  xx = w + kw - 1;
      const bool ok = ((unsigned)y < (unsigned)Hdim) &&
                      ((unsigned)xx < (unsigned)Wdim);
      const int  yc = ok ? y  : 0;       // clamp -> always in-bounds address
      const int  xc = ok ? xx : 0;
      const float* src = xb + yc * Wdim + xc;           // + cin*HW
      // LDS slot for (tap, cin=wv, fn); cc advances cin by 8 => +256 floats
      float* dst = &lds[((tap * 32 + (wv >> 1)) * NTB + fn) * 2 + (wv & 1)];
      #pragma unroll
      for (int cc = 0; cc < 8; ++cc) {
        const int cin = cc * 8 + wv;
        float v = src[(size_t)cin * HWsz];              // imm-offset stream
        v = ok ? v : 0.0f;                              // padding -> zero
        dst[cc * (4 * NTB * 2)] = v;
      }
    }
  }
  __syncthreads();

  // ---------------- Phase 2: flat-K GEMM ------------------------------------
  const int mrow = wave * 16 + col;
  const float* Arow = wA + (size_t)mrow * KTOT;
  const v2f*   ldsp = (const v2f*)lds;

  v8f acc0 = {};   // pixels n0 + [0,16)
  v8f acc1 = {};   // pixels n0 + [16,32)
  #pragma unroll 4
  for (int ks = 0; ks < KTOT / 4; ++ks) {    // 144 WMMA steps
    v2f a;
    if (PACKED) {
      a = *(const v2f*)(Arow + ks * 4 + khalf * 2);   // contiguous b64 stream
    } else {
      const int c0  = ks * 4 + khalf * 2;   // even; c0&63 <= 62 so same tap
      const int cin = c0 & 63;
      const int tap = c0 >> 6;
      a.x = Arow[cin * 9 + tap];
      a.y = Arow[(cin + 1) * 9 + tap];
    }
    const v2f* Bp = ldsp + (ks * 2 + khalf) * NTB;
    v2f b0 = Bp[col];
    v2f b1 = Bp[col + 16];
    acc0 = __builtin_amdgcn_wmma_f32_16x16x4_f32(
        false, a, false, b0, (short)0, acc0, false, false);
    acc1 = __builtin_amdgcn_wmma_f32_16x16x4_f32(
        false, a, false, b1, (short)0, acc1, false, false);
  }

  // ---------------- Epilogue: bias + store (no divisions) -------------------
  // C/D layout: VGPR v -> row (wave*16 + khalf*8 + v), column = lane&15.
  const int hwA = hw0 + col;
  const int hwB = hw0 + 16 + col;
  float* outb = out + (size_t)bb * (COUT * HWsz);
  #pragma unroll
  for (int v = 0; v < 8; ++v) {
    const int r = wave * 16 + khalf * 8 + v;
    const float bs = bias[r];
    outb[(size_t)r * HWsz + hwA] = acc0[v] + bs;
    outb[(size_t)r * HWsz + hwB] = acc1[v] + bs;
  }
}

extern "C" void kernel_launch(void* const* d_in, const int* in_sizes, int n_in,
                              void* d_out, int out_size, void* d_ws, size_t ws_size,
                              hipStream_t stream) {
  const float* x    = (const float*)d_in[0];
  const float* wgt  = (const float*)d_in[1];
  const float* bias = (const float*)d_in[2];
  float* out = (float*)d_out;

  const size_t packBytes = PACK_ELEMS * sizeof(float);  // 294912
  if (ws_size >= packBytes) {
    float* wp = (float*)d_ws;
    pack_weights<<<dim3(PACK_ELEMS / 256), dim3(256), 0, stream>>>(wgt, wp);
    conv3x3_wmma<true><<<dim3(Npix / NTB), dim3(256), 0, stream>>>(x, wp, bias, out);
  } else {
    conv3x3_wmma<false><<<dim3(Npix / NTB), dim3(256), 0, stream>>>(x, wgt, bias, out);
  }
}